// MeanStiffRegularizer_43104291782997
// MI455X (gfx1250) — compile-verified
//
#include <hip/hip_runtime.h>

#define BINS 256

typedef __attribute__((ext_vector_type(2))) float v2f;
typedef __attribute__((ext_vector_type(8))) float v8f;
typedef __attribute__((ext_vector_type(4))) float v4f;
typedef __attribute__((ext_vector_type(4))) int   v4i;

// ---------------------------------------------------------------------------
// Kernel 0: zero the 512-float global accumulator (sums[256] ++ counts[256])
// Must run every call: harness poisons d_ws once and never re-poisons.
// ---------------------------------------------------------------------------
__global__ __launch_bounds__(256) void MSR_zero_ws(float* __restrict__ g) {
    int t = blockIdx.x * blockDim.x + threadIdx.x;
    if (t < 2 * BINS) g[t] = 0.0f;
}

// ---------------------------------------------------------------------------
// Kernel 1: streaming histogram. 256 MiB read once -> HBM-bound (~11.5us
// floor at 23.3 TB/s). Non-temporal 128-bit loads (data is never reused,
// working set > 192MB L2), LDS ds_add_f32 atomics for sum+count bins,
// per-block drain via global f32 atomics into d_ws.
// ---------------------------------------------------------------------------
__global__ __launch_bounds__(256) void MSR_hist(const float* __restrict__ x,
                                                const int*   __restrict__ idx,
                                                float*       __restrict__ gbins,
                                                int n) {
    __shared__ float lbins[2 * BINS];   // [0,256): sums, [256,512): counts
    for (int t = threadIdx.x; t < 2 * BINS; t += blockDim.x) lbins[t] = 0.0f;
    __syncthreads();

    const int tid    = blockIdx.x * blockDim.x + threadIdx.x;
    const int stride = gridDim.x * blockDim.x;
    const int nvec   = n >> 2;

    const v4f* __restrict__ x4 = (const v4f*)x;
    const v4i* __restrict__ i4 = (const v4i*)idx;

    for (int i = tid; i < nvec; i += stride) {
        if (i + stride < nvec) {
            __builtin_prefetch(&x4[i + stride], 0, 0);  // global_prefetch_b8
            __builtin_prefetch(&i4[i + stride], 0, 0);
        }
        v4f xv = __builtin_nontemporal_load(&x4[i]);
        v4i iv = __builtin_nontemporal_load(&i4[i]);

        atomicAdd(&lbins[iv.x], xv.x);  atomicAdd(&lbins[BINS + iv.x], 1.0f);
        atomicAdd(&lbins[iv.y], xv.y);  atomicAdd(&lbins[BINS + iv.y], 1.0f);
        atomicAdd(&lbins[iv.z], xv.z);  atomicAdd(&lbins[BINS + iv.z], 1.0f);
        atomicAdd(&lbins[iv.w], xv.w);  atomicAdd(&lbins[BINS + iv.w], 1.0f);
    }
    // scalar tail (n is 2^25 here, but stay generic)
    for (int i = (nvec << 2) + tid; i < n; i += stride) {
        float xv = x[i];
        int   b  = idx[i];
        atomicAdd(&lbins[b], xv);
        atomicAdd(&lbins[BINS + b], 1.0f);
    }

    __syncthreads();
    for (int t = threadIdx.x; t < 2 * BINS; t += blockDim.x)
        atomicAdd(&gbins[t], lbins[t]);
}

// ---------------------------------------------------------------------------
// Kernel 2: finalize with V_WMMA_F32_16X16X4_F32 (single wave32, EXEC all 1s).
// loss = 0.01 * mean(dev^2), dev[j] = sum[j]/max(cnt[j],1) - target[j].
// Sum of squares over 256 values = trace(A A^T) accumulated over 4 chunks of
// 64: A[m,k] = dev[base + k*16 + m], B = A^T, diag(D) holds row squared-sums.
// Full f32 precision (matches the f32 reference; no f16 quantization).
// ---------------------------------------------------------------------------
__global__ __launch_bounds__(32) void MSR_finalize(const float* __restrict__ gbins,
                                                   const float* __restrict__ tgt,
                                                   float*       __restrict__ out) {
    const int lane = threadIdx.x;
    const int col  = lane & 15;
    const int half = lane >> 4;   // 0 for lanes 0-15, 1 for lanes 16-31

    v8f acc = {0.f, 0.f, 0.f, 0.f, 0.f, 0.f, 0.f, 0.f};

    for (int c = 0; c < 4; ++c) {
        const int base = c * 64;
        auto dev = [&](int j) -> float {
            float s   = gbins[j];
            float cnt = gbins[BINS + j];
            float m   = s / fmaxf(cnt, 1.0f);   // empty segment -> 0/1 = 0
            return m - tgt[j];
        };
        // A 16x4 f32: lane holds M=col; VGPR0 K=(half?2:0), VGPR1 K=(half?3:1)
        v2f A, B;
        A.x = dev(base + (half * 2 + 0) * 16 + col);
        A.y = dev(base + (half * 2 + 1) * 16 + col);
        // B 4x16 f32 = A^T: lane holds N=col; VGPR0 K=(half?1:0), VGPR1 K=(half?3:2)
        B.x = dev(base + (half + 0) * 16 + col);
        B.y = dev(base + (half + 2) * 16 + col);

        // D = A*B + C  ->  v_wmma_f32_16x16x4_f32
        acc = __builtin_amdgcn_wmma_f32_16x16x4_f32(
            /*neg_a=*/false, A, /*neg_b=*/false, B,
            /*c_mod=*/(short)0, acc, /*reuse_a=*/false, /*reuse_b=*/false);
    }

    // Diagonal of 16x16 f32 D: m<8 -> VGPR m, lane m; m>=8 -> VGPR m-8, lane m+16.
    float d = 0.0f;
    if (lane < 8)        d = acc[lane];        // D[lane,lane]
    else if (lane >= 24) d = acc[lane - 24];   // D[lane-16, lane-16]

    // wave32 reduce
    for (int off = 16; off > 0; off >>= 1)
        d += __shfl_down(d, off, 32);

    if (lane == 0) out[0] = d * (0.01f / 256.0f);
}

// ---------------------------------------------------------------------------
extern "C" void kernel_launch(void* const* d_in, const int* in_sizes, int n_in,
                              void* d_out, int out_size, void* d_ws, size_t ws_size,
                              hipStream_t stream) {
    const float* x   = (const float*)d_in[0];
    const int*   idx = (const int*)  d_in[1];
    const float* tgt = (const float*)d_in[2];
    float*       out = (float*)d_out;
    float*       gbins = (float*)d_ws;   // 512 floats of scratch

    const int n = in_sizes[0];

    MSR_zero_ws<<<(2 * BINS + 255) / 256, 256, 0, stream>>>(gbins);

    const int threads = 256;
    const int blocks  = 1024;  // ~262k threads, ~32 float4 iters each; keeps
                               // global-atomic drain (1024*512 ops) cheap
    MSR_hist<<<blocks, threads, 0, stream>>>(x, idx, gbins, n);

    MSR_finalize<<<1, 32, 0, stream>>>(gbins, tgt, out);
}